// GCNLayer_31920196943930
// MI455X (gfx1250) — compile-verified
//
#include <hip/hip_runtime.h>
#include <stdint.h>

typedef __attribute__((ext_vector_type(2))) float v2f;
typedef __attribute__((ext_vector_type(8))) float v8f;

#define IN_C  256
#define OUT_C 128
#define A_PITCH 260   // floats per LDS row (pad 256->260 for conflict-free banks)

// ---------------- degree accumulation ----------------
__global__ void deg_init_kernel(float* __restrict__ deg, int n) {
    int i = blockIdx.x * blockDim.x + threadIdx.x;
    if (i < n) deg[i] = 1.0f;  // self-loop weight
}

__global__ void deg_acc_kernel(const long long* __restrict__ dst,
                               const float* __restrict__ w,
                               float* __restrict__ deg, int e) {
    int i = blockIdx.x * blockDim.x + threadIdx.x;
    if (i < e) {
        unsafeAtomicAdd(&deg[dst[i]], w[i]);   // -> global_atomic_add_f32
    }
}

// deg -> deg^-1/2 in place
__global__ void dinv_kernel(float* __restrict__ deg, int n) {
    int i = blockIdx.x * blockDim.x + threadIdx.x;
    if (i < n) {
        float d = deg[i];
        deg[i] = (d > 0.0f) ? rsqrtf(d) : 0.0f;
    }
}

// ---------------- W -> W^T (so B-operand pairs are contiguous) ----------------
__global__ void transpose_W_kernel(const float* __restrict__ W,
                                   float* __restrict__ Wt) {
    int idx = blockIdx.x * blockDim.x + threadIdx.x;   // 256*128 threads
    if (idx < IN_C * OUT_C) {
        int k = idx >> 7;          // row in W
        int c = idx & (OUT_C - 1); // col in W
        Wt[c * IN_C + k] = W[idx];
    }
}

// ---------------- GEMM: h = x @ W via fp32 WMMA ----------------
// Block = 256 threads = 8 waves. Block b covers rows [16b, 16b+16);
// wave w covers cols [16w, 16w+16). K marched in steps of 4.
// A tile (16 x 256 f32 = 16KB) is staged ONCE per block into padded LDS via
// the CDNA5 async global->LDS path, then all 8 waves read it conflict-free.
__global__ void __launch_bounds__(256)
gemm_h_kernel(const float* __restrict__ x,
              const float* __restrict__ Wt,  // [OUT_C, IN_C]
              float* __restrict__ h, int n) {
    __shared__ float Atile[16 * A_PITCH];

    const int tid  = threadIdx.x;
    const int wave = tid >> 5;
    const int lane = tid & 31;
    const int half = lane >> 4;       // 0: lanes 0-15 (K 0,1), 1: lanes 16-31 (K 2,3)
    const int l    = lane & 15;
    const int row0 = blockIdx.x * 16;

    // ---- async copy: 1024 x 16B chunks, coalesced; 4 chunks per thread ----
#pragma unroll
    for (int i = 0; i < 4; ++i) {
        int idx = tid + 256 * i;            // chunk id 0..1023
        int r   = idx >> 6;                 // source row in tile (64 chunks/row)
        int c   = (idx & 63) << 4;          // byte offset within row
        int rowg = row0 + r;
        if (rowg >= n) rowg = n - 1;        // clamp (dup rows never stored)
        uint64_t gaddr = (uint64_t)(uintptr_t)(x + (size_t)rowg * IN_C) + (uint64_t)c;
        uint32_t laddr = (uint32_t)(uintptr_t)(&Atile[0]) + (uint32_t)(r * (A_PITCH * 4) + c);
        asm volatile("global_load_async_to_lds_b128 %0, %1, off"
                     :: "v"(laddr), "v"(gaddr) : "memory");
    }
    asm volatile("s_wait_asynccnt 0x0" ::: "memory");
    __syncthreads();

    const int col = wave * 16 + l;
    // B pairs: Wt[col][2p..2p+1]; pair index p = k/2 + half
    const v2f* __restrict__ bcol = (const v2f*)(Wt + (size_t)col * IN_C);
    // A pairs from LDS: row l, element k + 2*half (pitch 260 -> banks 4l..4l+3)
    const float* arow = &Atile[l * A_PITCH + 2 * half];

    v8f acc = {};
#pragma unroll 8
    for (int k = 0; k < IN_C; k += 4) {
        v2f a = *(const v2f*)(arow + k);           // ds_load_b64, conflict-free
        v2f b = bcol[(k >> 1) + half];             // global_load_b64 (L2-resident)
        acc = __builtin_amdgcn_wmma_f32_16x16x4_f32(
            false, a, false, b, (short)0, acc, false, false);
    }

    // D 16x16 f32: VGPR r holds M = r (lanes 0-15) / M = r+8 (lanes 16-31)
    if (row0 + 16 <= n) {                          // uniform fast path
#pragma unroll
        for (int r = 0; r < 8; ++r)
            h[(size_t)(row0 + r + 8 * half) * OUT_C + col] = acc[r];
    } else {
#pragma unroll
        for (int r = 0; r < 8; ++r) {
            int m = row0 + r + 8 * half;
            if (m < n) h[(size_t)m * OUT_C + col] = acc[r];
        }
    }
}

// ---------------- init out with self-loop message ----------------
// out[n][c] = h[n][c] * dinv[n]^2  (writes every output element each launch)
__global__ void selfloop_kernel(const float* __restrict__ h,
                                const float* __restrict__ dinv,
                                float* __restrict__ out, int n) {
    int idx = blockIdx.x * blockDim.x + threadIdx.x;
    if (idx < n * OUT_C) {
        int node = idx >> 7;  // /128
        float di = dinv[node];
        out[idx] = h[idx] * di * di;
    }
}

// ---------------- edge scatter: out[dst] += h[src] * norm ----------------
// One wave per edge; lane owns 4 contiguous channels (float4 gather).
__global__ void scatter_kernel(const long long* __restrict__ src,
                               const long long* __restrict__ dst,
                               const float* __restrict__ w,
                               const float* __restrict__ dinv,
                               const float* __restrict__ h,
                               float* __restrict__ out, int e) {
    int edge = (blockIdx.x * blockDim.x + threadIdx.x) >> 5;
    int lane = threadIdx.x & 31;
    if (edge >= e) return;

    long long s = src[edge];
    long long d = dst[edge];
    float norm = dinv[s] * w[edge] * dinv[d];

    const float4* hv = (const float4*)(h + s * OUT_C);
    float4 v = hv[lane];                       // coalesced 512B per wave
    float* o = out + d * OUT_C + lane * 4;
    unsafeAtomicAdd(o + 0, v.x * norm);
    unsafeAtomicAdd(o + 1, v.y * norm);
    unsafeAtomicAdd(o + 2, v.z * norm);
    unsafeAtomicAdd(o + 3, v.w * norm);
}

// ---------------- bias + ReLU ----------------
__global__ void finalize_kernel(float* __restrict__ out,
                                const float* __restrict__ b, int n) {
    int idx = blockIdx.x * blockDim.x + threadIdx.x;
    if (idx < n * OUT_C) {
        int c = idx & (OUT_C - 1);
        out[idx] = fmaxf(out[idx] + b[c], 0.0f);
    }
}

extern "C" void kernel_launch(void* const* d_in, const int* in_sizes, int n_in,
                              void* d_out, int out_size, void* d_ws, size_t ws_size,
                              hipStream_t stream) {
    const float*      x  = (const float*)d_in[0];                 // [N, 256]
    const long long*  ei = (const long long*)d_in[1];             // [2, E] int64
    const float*      ew = (const float*)d_in[2];                 // [E]
    const float*      W  = (const float*)d_in[3];                 // [256, 128]
    const float*      b  = (const float*)d_in[4];                 // [128]
    float*            out = (float*)d_out;                        // [N, 128]

    const int N = in_sizes[0] / IN_C;
    const int E = in_sizes[2];
    const long long* src = ei;        // edge_index[0]
    const long long* dst = ei + E;    // edge_index[1]

    // workspace layout: h [N*128 f32] | deg/dinv [N f32] | Wt [128*256 f32]
    float* h   = (float*)d_ws;
    float* deg = h + (size_t)N * OUT_C;
    float* Wt  = deg + (size_t)N;

    const int T = 256;
    deg_init_kernel<<<(N + T - 1) / T, T, 0, stream>>>(deg, N);
    deg_acc_kernel<<<(E + T - 1) / T, T, 0, stream>>>(dst, ew, deg, E);
    dinv_kernel<<<(N + T - 1) / T, T, 0, stream>>>(deg, N);

    transpose_W_kernel<<<(IN_C * OUT_C + T - 1) / T, T, 0, stream>>>(W, Wt);

    gemm_h_kernel<<<(N + 15) / 16, T, 0, stream>>>(x, Wt, h, N);

    selfloop_kernel<<<(N * OUT_C + T - 1) / T, T, 0, stream>>>(h, deg, out, N);

    long long threads = (long long)E * 32;
    scatter_kernel<<<(unsigned)((threads + T - 1) / T), T, 0, stream>>>(
        src, dst, ew, deg, h, out, E);

    finalize_kernel<<<(N * OUT_C + T - 1) / T, T, 0, stream>>>(out, b, N);
}